// SGC_LL_64510408786278
// MI455X (gfx1250) — compile-verified
//
#include <hip/hip_runtime.h>
#include <hip/hip_bf16.h>
#include <math.h>

// SGC_LL on MI455X (gfx1250): all dense stages via V_WMMA_F32_16X16X4_F32
// (full-f32 semantics, f32 accumulate).
// Elementwise stages (RBF affinity, degree normalization, Laplacian) are
// wave32 VALU + shuffle reductions.

#define BATCH 256
#define NATOM 256
#define FIN   128
#define KCH   4
#define NFILT 256
#define FK    (FIN*KCH)   // 512

typedef float v2f __attribute__((ext_vector_type(2)));
typedef float v8f __attribute__((ext_vector_type(8)));

// ---- WMMA f32 16x16x4 tile helpers (layouts per CDNA5 ISA 7.12.2) ----
// A 16x4 (MxK): lanes 0-15 -> M=lane, VGPR0/1 = K0,K1; lanes 16-31 -> K2,K3.
__device__ __forceinline__ v2f wmma_load_a(const float* src, int ld) {
  int l  = threadIdx.x & 31;
  int m  = l & 15;
  int kk = (l >> 4) << 1;           // 0 or 2
  v2f a;
  a.x = src[m * ld + kk];
  a.y = src[m * ld + kk + 1];
  return a;
}
// B 4x16 (KxN) from row-major src: lanes 0-15 -> N=lane, rows {0,1}; 16-31 -> rows {2,3}.
__device__ __forceinline__ v2f wmma_load_b(const float* src, int ld) {
  int l  = threadIdx.x & 31;
  int n  = l & 15;
  int kk = (l >> 4) << 1;
  v2f b;
  b.x = src[kk * ld + n];
  b.y = src[(kk + 1) * ld + n];
  return b;
}
// B 4x16 with generic strides: B[k][n] = src[k*kstride + n*nstride]
__device__ __forceinline__ v2f wmma_load_b_strided(const float* src, int kstride, int nstride) {
  int l  = threadIdx.x & 31;
  int n  = l & 15;
  int kk = (l >> 4) << 1;
  v2f b;
  b.x = src[kk * kstride + n * nstride];
  b.y = src[(kk + 1) * kstride + n * nstride];
  return b;
}
__device__ __forceinline__ v8f wmma_f32(v2f a, v2f b, v8f c) {
  // 8 args: (neg_a, A, neg_b, B, c_mod, C, reuse_a, reuse_b)
  return __builtin_amdgcn_wmma_f32_16x16x4_f32(false, a, false, b, (short)0, c, false, false);
}

// ---- Stage 1: xw = (x * mask) @ M_L ; also scatter masked x into xc slot k=0 ----
__global__ void k_xw(const float* __restrict__ x, const int* __restrict__ n_atoms,
                     const float* __restrict__ M_L, float* __restrict__ xw,
                     float* __restrict__ xc) {
  __shared__ float sA[16 * FIN];     // 8 KB masked-x panel
  int b  = blockIdx.y;
  int m0 = blockIdx.x * 16;
  int na = n_atoms[b];
  for (int i = threadIdx.x; i < 16 * FIN; i += blockDim.x) {
    int m = i >> 7, f = i & 127;
    int row = m0 + m;
    float v = (row < na) ? x[((size_t)b * NATOM + row) * FIN + f] : 0.f;
    sA[i] = v;
    xc[((size_t)b * NATOM + row) * FK + f * KCH + 0] = v;   // x0 slot
  }
  __syncthreads();
  int wave = threadIdx.x >> 5;
  int n0 = wave * 16;                // 8 waves cover F=128 output cols
  v8f acc = {};
  for (int k0 = 0; k0 < FIN; k0 += 4) {
    v2f a  = wmma_load_a(sA + k0, FIN);
    v2f bm = wmma_load_b(M_L + (size_t)k0 * FIN + n0, FIN);
    acc = wmma_f32(a, bm, acc);
  }
  int l  = threadIdx.x & 31;
  int n  = n0 + (l & 15);
  int mb = m0 + ((l >> 4) << 3);
#pragma unroll
  for (int r = 0; r < 8; ++r)
    xw[((size_t)b * NATOM + mb + r) * FIN + n] = acc[r];
}

// ---- Stage 2: sq[b,i] = sum_f xw^2 (one row per wave, shuffle reduce) ----
__global__ void k_sq(const float* __restrict__ xw, float* __restrict__ sq) {
  int wave = threadIdx.x >> 5;
  int l    = threadIdx.x & 31;
  int row  = blockIdx.x * 4 + wave;          // [0, B*N)
  const float* p = xw + (size_t)row * FIN;
  float s = 0.f;
  for (int f = l; f < FIN; f += 32) { float v = p[f]; s += v * v; }
  for (int off = 16; off > 0; off >>= 1) s += __shfl_xor(s, off, 32);
  if (l == 0) sq[row] = s;
}

// ---- Stage 3: Gram via WMMA -> d2 -> W = exp(-sqrt(max(d2,1e-12)))*off ----
__global__ void k_gram_w(const float* __restrict__ xw, const float* __restrict__ sq,
                         const int* __restrict__ n_atoms, float* __restrict__ Wb) {
  __shared__ float sA[16 * FIN];
  int b  = blockIdx.y;
  int m0 = blockIdx.x * 16;
  int na = n_atoms[b];
  const float* xwb = xw + (size_t)b * NATOM * FIN;
  for (int i = threadIdx.x; i < 16 * FIN; i += blockDim.x)
    sA[i] = xwb[(size_t)(m0 + (i >> 7)) * FIN + (i & 127)];
  __syncthreads();
  int wave = threadIdx.x >> 5;
  int l    = threadIdx.x & 31;
  for (int t = 0; t < 2; ++t) {              // 8 waves x 2 = 16 col tiles (N=256)
    int n0 = (wave * 2 + t) * 16;
    v8f acc = {};
    for (int k0 = 0; k0 < FIN; k0 += 4) {
      v2f a  = wmma_load_a(sA + k0, FIN);
      v2f bm = wmma_load_a(xwb + (size_t)n0 * FIN + k0, FIN);  // B = xw^T tile
      acc = wmma_f32(a, bm, acc);
    }
    int n  = n0 + (l & 15);
    int mb = m0 + ((l >> 4) << 3);
    float sqn = sq[(size_t)b * NATOM + n];
#pragma unroll
    for (int r = 0; r < 8; ++r) {
      int m = mb + r;
      float d2 = sq[(size_t)b * NATOM + m] + sqn - 2.f * acc[r];
      bool offd = (m != n) && (m < na) && (n < na);
      float w = offd ? expf(-sqrtf(fmaxf(d2, 1e-12f))) : 0.f;
      Wb[((size_t)b * NATOM + m) * NATOM + n] = w;
    }
  }
}

// ---- Stage 4: deg = rowsum(W) + spacing(0); dinv = mask ? rsqrt : 0 ----
__global__ void k_deg(const float* __restrict__ Wb, const int* __restrict__ n_atoms,
                      float* __restrict__ dinv) {
  int wave = threadIdx.x >> 5;
  int l    = threadIdx.x & 31;
  int row  = blockIdx.x * 4 + wave;
  int b = row >> 8, i = row & 255;
  const float* p = Wb + (size_t)row * NATOM;
  float s = 0.f;
  for (int j = l; j < NATOM; j += 32) s += p[j];
  for (int off = 16; off > 0; off >>= 1) s += __shfl_xor(s, off, 32);
  s += 1.401298464324817e-45f;               // np.spacing(np.float32(0))
  float dv = (i < n_atoms[b]) ? (1.f / sqrtf(s)) : 0.f;
  if (l == 0) dinv[row] = dv;
}

// ---- Stage 5: L_learn -> d_out; L = L_learn + lap_in*mask_i*mask_j -> Wb (in place) ----
__global__ void k_L(const float* __restrict__ lap_in, const int* __restrict__ n_atoms,
                    const float* __restrict__ dinv, float* __restrict__ Wb,
                    float* __restrict__ Lout) {
  size_t idx = (size_t)blockIdx.x * blockDim.x + threadIdx.x;   // [0, B*N*N)
  int b   = (int)(idx >> 16);
  int rem = (int)(idx & 65535);
  int m = rem >> 8, n = rem & 255;
  int na = n_atoms[b];
  float mm = (m < na) ? 1.f : 0.f;
  float mn = (n < na) ? 1.f : 0.f;
  float w  = Wb[idx];
  float ll = ((m == n) ? mm : 0.f)
           - dinv[(size_t)b * NATOM + m] * w * dinv[(size_t)b * NATOM + n];
  Lout[idx] = ll;
  Wb[idx] = ll + lap_in[idx] * mm * mn;
}

// ---- Stage 6: Chebyshev step x_dst = alpha * L @ x_cur - (do_sub ? x_prev : 0) ----
__global__ void k_cheb(const float* __restrict__ Lb, float* __restrict__ xc,
                       int kcur, int kprev, int kdst, float alpha, int do_sub) {
  __shared__ float sA[16 * NATOM];           // 16 KB L-row panel
  int b  = blockIdx.y;
  int m0 = blockIdx.x * 16;
  const float* Lrow = Lb + ((size_t)b * NATOM + m0) * NATOM;
  for (int i = threadIdx.x; i < 16 * NATOM; i += blockDim.x) sA[i] = Lrow[i];
  __syncthreads();
  int wave = threadIdx.x >> 5;
  int f0 = wave * 16;                        // 8 waves cover F=128
  v8f acc = {};
  for (int k0 = 0; k0 < NATOM; k0 += 4) {
    v2f a  = wmma_load_a(sA + k0, NATOM);
    // B[k][f] = xc[b, k0+k, f*K + kcur]
    v2f bm = wmma_load_b_strided(xc + ((size_t)b * NATOM + k0) * FK + f0 * KCH + kcur,
                                 FK, KCH);
    acc = wmma_f32(a, bm, acc);
  }
  int l  = threadIdx.x & 31;
  int n  = f0 + (l & 15);
  int mb = m0 + ((l >> 4) << 3);
#pragma unroll
  for (int r = 0; r < 8; ++r) {
    size_t base = ((size_t)b * NATOM + mb + r) * FK + n * KCH;
    float v = alpha * acc[r];
    if (do_sub) v -= xc[base + kprev];
    xc[base + kdst] = v;
  }
}

// ---- Stage 7: out = relu((xc @ weight + bias) * mask) ----
__global__ void k_out(const float* __restrict__ xc, const float* __restrict__ weight,
                      const float* __restrict__ bias, const int* __restrict__ n_atoms,
                      float* __restrict__ out) {
  __shared__ float sA[16 * FK];              // 32 KB xc panel
  int b  = blockIdx.y;
  int m0 = blockIdx.x * 16;
  int na = n_atoms[b];
  const float* xcr = xc + ((size_t)b * NATOM + m0) * FK;
  for (int i = threadIdx.x; i < 16 * FK; i += blockDim.x) sA[i] = xcr[i];
  __syncthreads();
  int wave = threadIdx.x >> 5;
  int l    = threadIdx.x & 31;
  for (int t = 0; t < 2; ++t) {              // 8 waves x 2 = 16 col tiles (FILT=256)
    int n0 = (wave * 2 + t) * 16;
    v8f acc = {};
    for (int k0 = 0; k0 < FK; k0 += 4) {
      v2f a  = wmma_load_a(sA + k0, FK);
      v2f bm = wmma_load_b(weight + (size_t)k0 * NFILT + n0, NFILT);
      acc = wmma_f32(a, bm, acc);
    }
    int n  = n0 + (l & 15);
    int mb = m0 + ((l >> 4) << 3);
    float bs = bias[n];
#pragma unroll
    for (int r = 0; r < 8; ++r) {
      int m = mb + r;
      float v = acc[r] + bs;
      v = (m < na) ? v : 0.f;
      out[((size_t)b * NATOM + m) * NFILT + n] = fmaxf(v, 0.f);
    }
  }
}

extern "C" void kernel_launch(void* const* d_in, const int* in_sizes, int n_in,
                              void* d_out, int out_size, void* d_ws, size_t ws_size,
                              hipStream_t stream) {
  const float* x       = (const float*)d_in[0];
  const float* lap_in  = (const float*)d_in[1];
  const int*   n_atoms = (const int*)d_in[2];
  const float* M_L     = (const float*)d_in[3];
  const float* weight  = (const float*)d_in[4];
  const float* bias    = (const float*)d_in[5];

  float* out  = (float*)d_out;                               // [B,N,FILT]
  float* Lout = out + (size_t)BATCH * NATOM * NFILT;         // [B,N,N] (L_learn)

  // Workspace layout (~235 MB of f32):
  float* ws   = (float*)d_ws;
  float* xw   = ws;                                          // B*N*F
  float* sq   = xw   + (size_t)BATCH * NATOM * FIN;          // B*N
  float* dinv = sq   + (size_t)BATCH * NATOM;                // B*N
  float* Wb   = dinv + (size_t)BATCH * NATOM;                // B*N*N (W, then L in place)
  float* xc   = Wb   + (size_t)BATCH * NATOM * NATOM;        // B*N*F*K

  dim3 gT(NATOM / 16, BATCH);                                // 16 row-tiles x 256 mols
  k_xw    <<<gT, 256, 0, stream>>>(x, n_atoms, M_L, xw, xc);
  k_sq    <<<(BATCH * NATOM) / 4, 128, 0, stream>>>(xw, sq);
  k_gram_w<<<gT, 256, 0, stream>>>(xw, sq, n_atoms, Wb);
  k_deg   <<<(BATCH * NATOM) / 4, 128, 0, stream>>>(Wb, n_atoms, dinv);
  k_L     <<<(BATCH * NATOM * NATOM) / 256, 256, 0, stream>>>(lap_in, n_atoms, dinv, Wb, Lout);
  k_cheb  <<<gT, 256, 0, stream>>>(Wb, xc, 0, 0, 1, 1.0f, 0);   // x1 = L x0
  k_cheb  <<<gT, 256, 0, stream>>>(Wb, xc, 1, 0, 2, 2.0f, 1);   // x2 = 2 L x1 - x0
  k_cheb  <<<gT, 256, 0, stream>>>(Wb, xc, 2, 1, 3, 2.0f, 1);   // x3 = 2 L x2 - x1
  k_out   <<<gT, 256, 0, stream>>>(xc, weight, bias, n_atoms, out);
}